// YOLOv11Loss_29858612642435
// MI455X (gfx1250) — compile-verified
//
#include <hip/hip_runtime.h>
#include <math.h>

#define NCLS   80
#define NB     32
#define NTGT   400
#define CAP    2048
#define REC    (NCLS + 5)   // 85 floats per prediction record

typedef float v2f __attribute__((ext_vector_type(2)));
typedef float v8f __attribute__((ext_vector_type(8)));

__device__ __forceinline__ float softplusf(float x) {
  // stable: max(x,0) + log1p(exp(-|x|))  (matches jax.nn.softplus)
  return fmaxf(x, 0.0f) + log1pf(expf(-fabsf(x)));
}
__device__ __forceinline__ float sigmoidf(float x) {
  return 1.0f / (1.0f + expf(-x));
}

// ---------------------------------------------------------------------------
// Sum (a0+a1) over all 32 lanes of a wave with ONE fp32 WMMA:
// A(16x4) = the wave's 64 partials (2 VGPRs x 32 lanes), B(4x16) = ones, so
// D[m,n] = a0_m + a1_m + a0_{m+16} + a1_{m+16} for every n. Lane L holds
// D rows 8*(L/16)..8*(L/16)+7 in its 8 D VGPRs; fold those, then xor-16.
// Requires EXEC == all ones: call only from fully converged code.
// ---------------------------------------------------------------------------
__device__ __forceinline__ float wave_sum32(float a0, float a1) {
  v2f A;     A.x = a0;   A.y = a1;
  v2f Bones; Bones.x = 1.0f; Bones.y = 1.0f;
  v8f C = {0.f, 0.f, 0.f, 0.f, 0.f, 0.f, 0.f, 0.f};
  v8f D = __builtin_amdgcn_wmma_f32_16x16x4_f32(false, A, false, Bones,
                                                (short)0, C, false, false);
  float s = ((D[0] + D[1]) + (D[2] + D[3])) + ((D[4] + D[5]) + (D[6] + D[7]));
  s += __shfl_xor(s, 16, 32);
  return s;
}

// Block-wide (blockDim.x <= 256, multiple of 32) sum -> atomicAdd(dst).
// All threads of the block must reach this (predicate work, don't branch out).
__device__ __forceinline__ void block_sum_atomic(float a0, float a1,
                                                 float* dst, float* red) {
  float w  = wave_sum32(a0, a1);
  int wave = threadIdx.x >> 5;
  int lane = threadIdx.x & 31;
  if (lane == 0) red[wave] = w;
  __syncthreads();
  if (wave == 0) {                       // uniform per wave -> EXEC full
    int nw  = (int)(blockDim.x >> 5);
    float t = (lane < nw) ? red[lane] : 0.0f;
    float s = wave_sum32(t, 0.0f);
    if (lane == 0) atomicAdd(dst, s);
  }
  __syncthreads();                       // allow red[] reuse by caller
}

// ---------------------------------------------------------------------------
// Kernel 1: build_targets (na==1). One thread per GT box, level = blockIdx.y.
// Appends matched entries {b,gj,gi,c, tbx,tby,tbw,tbh} via atomic counter.
// ---------------------------------------------------------------------------
__global__ void k_build(const float* __restrict__ tg,
                        int* __restrict__ ent, int* __restrict__ cnt) {
  int lvl = blockIdx.y;
  int t   = blockIdx.x * blockDim.x + threadIdx.x;
  if (t >= NTGT) return;

  const int   Wd  = 80 >> lvl;                 // 80, 40, 20 (H == W)
  const float str = 8.0f * (float)(1 << lvl);  // 8, 16, 32

  const float* r = tg + t * 6;
  float bb = r[0], cc = r[1];
  float gx = r[2] * (float)Wd, gy = r[3] * (float)Wd;
  float gw = r[4] * (float)Wd, gh = r[5] * (float)Wd;

  float rx = gw / str, ry = gh / str;
  float m  = fmaxf(fmaxf(rx, 1.0f / rx), fmaxf(ry, 1.0f / ry));
  if (!(m < 4.0f)) return;

  const float offx[5] = {0.f, 0.5f, 0.f, -0.5f, 0.f};
  const float offy[5] = {0.f, 0.f, 0.5f, 0.f, -0.5f};
  float ix = (float)Wd - gx, iy = (float)Wd - gy;
  bool cond[5];
  cond[0] = true;
  cond[1] = ((gx - floorf(gx)) < 0.5f) && (gx > 1.0f);
  cond[2] = ((gy - floorf(gy)) < 0.5f) && (gy > 1.0f);
  cond[3] = ((ix - floorf(ix)) < 0.5f) && (ix > 1.0f);
  cond[4] = ((iy - floorf(iy)) < 0.5f) && (iy > 1.0f);

  for (int o = 0; o < 5; ++o) {
    if (!cond[o]) continue;
    int gi = (int)floorf(gx - offx[o]); gi = min(max(gi, 0), Wd - 1);
    int gj = (int)floorf(gy - offy[o]); gj = min(max(gj, 0), Wd - 1);
    int slot = atomicAdd(&cnt[lvl], 1);
    if (slot >= CAP) continue;                 // capacity 5*NTGT <= CAP anyway
    int* e = ent + (size_t)(lvl * CAP + slot) * 8;
    e[0] = (int)bb; e[1] = gj; e[2] = gi; e[3] = (int)cc;
    e[4] = __float_as_int(gx - (float)gi);
    e[5] = __float_as_int(gy - (float)gj);
    e[6] = __float_as_int(gw);
    e[7] = __float_as_int(gh);
  }
}

// ---------------------------------------------------------------------------
// Kernel 2: per-match gather + CIoU (lbox), tobj scatter, class BCE (lcls).
// All lanes stay live (work predicated by `active`) so the WMMA reductions
// run with EXEC == all ones. Block-uniform tail guard only.
// ---------------------------------------------------------------------------
__global__ void k_match(const float* __restrict__ p0,
                        const float* __restrict__ p1,
                        const float* __restrict__ p2,
                        const int* __restrict__ ent,
                        const int* __restrict__ cnt,
                        float* __restrict__ tobj_ws,
                        float* __restrict__ acc) {  // [0..2]=lbox [3..5]=lcls
  __shared__ float red[8];
  const int lvl = blockIdx.y;
  int n = min(cnt[lvl], CAP);
  if ((int)(blockIdx.x * blockDim.x) >= n) return;       // uniform per block

  int  idx    = blockIdx.x * blockDim.x + threadIdx.x;
  bool active = idx < n;
  int  eidx   = active ? idx : 0;

  const int    Wd   = 80 >> lvl;
  const float  strf = 8.0f * (float)(1 << lvl);
  const float* p    = (lvl == 0) ? p0 : ((lvl == 1) ? p1 : p2);
  const size_t c0 = (size_t)NB * 80 * 80, c1 = (size_t)NB * 40 * 40;
  float* tobj = tobj_ws + ((lvl == 0) ? 0 : ((lvl == 1) ? c0 : c0 + c1));

  const int* e  = ent + (size_t)(lvl * CAP + eidx) * 8;
  int   b = e[0], gj = e[1], gi = e[2], tc = e[3];
  float tbx = __int_as_float(e[4]), tby = __int_as_float(e[5]);
  float tbw = __int_as_float(e[6]), tbh = __int_as_float(e[7]);

  size_t cell = ((size_t)b * Wd + gj) * Wd + gi;
  const float* row = p + cell * REC;

  float px = sigmoidf(row[0]) * 2.0f - 0.5f;
  float py = sigmoidf(row[1]) * 2.0f - 0.5f;
  float sw = sigmoidf(row[2]) * 2.0f;  float pw = sw * sw * strf;
  float sh = sigmoidf(row[3]) * 2.0f;  float ph = sh * sh * strf;

  // CIoU (matches bbox_iou(..., CIoU=True))
  const float eps = 1e-7f;
  float b1x1 = px - pw * 0.5f, b1x2 = px + pw * 0.5f;
  float b1y1 = py - ph * 0.5f, b1y2 = py + ph * 0.5f;
  float b2x1 = tbx - tbw * 0.5f, b2x2 = tbx + tbw * 0.5f;
  float b2y1 = tby - tbh * 0.5f, b2y2 = tby + tbh * 0.5f;
  float iw = fmaxf(fminf(b1x2, b2x2) - fmaxf(b1x1, b2x1), 0.0f);
  float ih = fmaxf(fminf(b1y2, b2y2) - fmaxf(b1y1, b2y1), 0.0f);
  float inter = iw * ih;
  float uni   = pw * ph + tbw * tbh - inter + eps;
  float iou   = inter / uni;
  float cw = fmaxf(b1x2, b2x2) - fminf(b1x1, b2x1);
  float chh = fmaxf(b1y2, b2y2) - fminf(b1y1, b2y1);
  float c2 = cw * cw + chh * chh + eps;
  float dx = b2x1 + b2x2 - b1x1 - b1x2;
  float dy = b2y1 + b2y2 - b1y1 - b1y2;
  float rho2 = (dx * dx + dy * dy) * 0.25f;
  float da = atanf(tbw / tbh) - atanf(pw / ph);
  float v  = 0.40528473456f * da * da;              // 4/pi^2
  float alpha = v / (v - iou + 1.0f + eps);
  float cio = iou - (rho2 / c2 + v * alpha);

  float lbox_c = active ? (1.0f - cio) : 0.0f;
  if (active) tobj[cell] = fmaxf(cio, 0.0f);        // .at[].set semantics

  float lcls_c = 0.0f;
  for (int k = 0; k < NCLS; ++k) lcls_c += softplusf(row[5 + k]);
  lcls_c -= row[5 + tc];                            // - sum(logit * onehot)
  if (!active) lcls_c = 0.0f;

  block_sum_atomic(lbox_c, 0.0f, acc + lvl,     red);
  block_sum_atomic(lcls_c, 0.0f, acc + 3 + lvl, red);
}

// ---------------------------------------------------------------------------
// Kernel 3: dense objectness BCE over all B*H*W cells of one level.
// Grid-stride streaming loop with two accumulators (feeds the 2 A-VGPRs of
// the fp32 WMMA reduction). p[..,4] is read at stride REC floats.
// ---------------------------------------------------------------------------
__global__ void k_obj(const float* __restrict__ p,
                      const float* __restrict__ tobj,
                      long N, float* __restrict__ dst) {
  __shared__ float red[8];
  float a0 = 0.0f, a1 = 0.0f;
  long gsz = (long)gridDim.x * blockDim.x;
  long i   = (long)blockIdx.x * blockDim.x + threadIdx.x;
  for (; i + gsz < N; i += 2 * gsz) {
    __builtin_prefetch(p + (i + 2 * gsz) * REC + 4, 0, 0);
    float x0 = p[i * REC + 4];
    float x1 = p[(i + gsz) * REC + 4];
    a0 += softplusf(x0) - x0 * tobj[i];
    a1 += softplusf(x1) - x1 * tobj[i + gsz];
  }
  if (i < N) {
    float x = p[i * REC + 4];
    a0 += softplusf(x) - x * tobj[i];
  }
  block_sum_atomic(a0, a1, dst, red);               // converged here
}

// ---------------------------------------------------------------------------
// Kernel 4: combine. acc: [0..2]=lbox_sum [3..5]=lcls_sum [6..8]=lobj_sum.
// out = [loss, lbox, lobj, lcls]
// ---------------------------------------------------------------------------
__global__ void k_final(const int* __restrict__ cnt,
                        const float* __restrict__ acc,
                        float* __restrict__ out) {
  const float bal[3]  = {4.0f, 1.0f, 0.4f};
  const float cells[3] = {(float)(NB * 80 * 80), (float)(NB * 40 * 40),
                          (float)(NB * 20 * 20)};
  float lbox = 0.0f, lobj = 0.0f, lcls = 0.0f;
  for (int i = 0; i < 3; ++i) {
    int n = min(cnt[i], CAP);
    if (n > 0) {
      lbox += acc[i]     / (float)n;
      lcls += acc[3 + i] / ((float)n * (float)NCLS);
    }
    lobj += (acc[6 + i] / cells[i]) * bal[i];
  }
  lbox *= 7.5f; lobj *= 1.0f; lcls *= 0.5f;
  out[0] = (lbox + lobj + lcls) * (float)NB;
  out[1] = lbox; out[2] = lobj; out[3] = lcls;
}

// ---------------------------------------------------------------------------
extern "C" void kernel_launch(void* const* d_in, const int* in_sizes, int n_in,
                              void* d_out, int out_size, void* d_ws,
                              size_t ws_size, hipStream_t stream) {
  const float* p0 = (const float*)d_in[0];
  const float* p1 = (const float*)d_in[1];
  const float* p2 = (const float*)d_in[2];
  const float* tg = (const float*)d_in[3];

  const size_t c0 = (size_t)NB * 80 * 80;   // 204800
  const size_t c1 = (size_t)NB * 40 * 40;   //  51200
  const size_t c2 = (size_t)NB * 20 * 20;   //  12800

  // Workspace layout (floats/ints): tobj grids | entries | counters | accums
  float* tobj = (float*)d_ws;
  int*   ent  = (int*)(tobj + c0 + c1 + c2);
  int*   cnt  = ent + (size_t)3 * CAP * 8;
  float* acc  = (float*)(cnt + 4);          // lbox[3], lcls[3], lobj[3]
  size_t zbytes = (size_t)((char*)(acc + 12) - (char*)d_ws);
  hipMemsetAsync(d_ws, 0, zbytes, stream);  // zero tobj + counters + accums

  k_build<<<dim3(1, 3), 512, 0, stream>>>(tg, ent, cnt);
  k_match<<<dim3(CAP / 256, 3), 256, 0, stream>>>(p0, p1, p2, ent, cnt, tobj,
                                                  acc);
  k_obj<<<200, 256, 0, stream>>>(p0, tobj,           (long)c0, acc + 6);
  k_obj<<<50,  256, 0, stream>>>(p1, tobj + c0,      (long)c1, acc + 7);
  k_obj<<<13,  256, 0, stream>>>(p2, tobj + c0 + c1, (long)c2, acc + 8);
  k_final<<<1, 1, 0, stream>>>(cnt, acc, (float*)d_out);
}